// LinkedFocalLoss_11982958755934
// MI455X (gfx1250) — compile-verified
//
#include <hip/hip_runtime.h>

// LinkedFocalLoss for MI455X (gfx1250, wave32).
// Kernel 1: one wave32 per row; streaming NT b128 loads; exact softmax + argmax
//           in registers; fast __expf (v_exp_f32 TRANS, co-executes with VALU)
//           for the 1000-term sum; per-row loss -> d_ws. Bandwidth bound: 524MB.
// Kernel 2: deterministic mean of 131072 floats using chained
//           V_WMMA_F32_16X16X4_F32 with an all-ones B matrix and 4 independent
//           accumulators to break the WMMA->WMMA RAW dependency chain.

#define N_ROWS 131072
#define C_DIM  1000
#define ALPHA_LINK 0.25f

typedef float v2f __attribute__((ext_vector_type(2)));
typedef float v4f __attribute__((ext_vector_type(4)));
typedef float v8f __attribute__((ext_vector_type(8)));

__global__ void __launch_bounds__(256)
lfl_rowloss(const float* __restrict__ y_pred,
            const int*   __restrict__ y_true,
            const int*   __restrict__ link_matrix,
            const int*   __restrict__ has_link,
            float*       __restrict__ row_loss) {
  const int lane = threadIdx.x & 31;
  const int wave = threadIdx.x >> 5;
  const int row  = blockIdx.x * 8 + wave;   // grid sized to N_ROWS/8 waves
  if (row >= N_ROWS) return;

  const float* rowp = y_pred + (size_t)row * C_DIM;   // 4000B rows -> 16B aligned
  const v4f*   rp4  = (const v4f*)rowp;
  const float  NEG_INF = -__builtin_inff();

  // 250 float4 chunks per row; lane handles chunks lane, lane+32, ... (8 iters).
  v4f buf[8];
#pragma unroll
  for (int it = 0; it < 8; ++it) {
    const int c = it * 32 + lane;
    if (c < 250) {
      buf[it] = __builtin_nontemporal_load(rp4 + c);   // stream, don't cache
    } else {
      v4f pad = {NEG_INF, NEG_INF, NEG_INF, NEG_INF};
      buf[it] = pad;
    }
  }

  // Local max + first-occurrence argmax over this lane's 32 values.
  float lm = NEG_INF;
  float bv = NEG_INF;
  int   bi = 0x7fffffff;
#pragma unroll
  for (int it = 0; it < 8; ++it) {
    const int base = (it * 32 + lane) * 4;
#pragma unroll
    for (int j = 0; j < 4; ++j) {
      const float x = buf[it][j];
      lm = fmaxf(lm, x);
      if (x > bv) { bv = x; bi = base + j; }   // strict > keeps lowest index
    }
  }

  // Wave-wide max (butterfly; wave32).
  float M = lm;
#pragma unroll
  for (int off = 16; off >= 1; off >>= 1)
    M = fmaxf(M, __shfl_xor(M, off, 32));

  // Wave-wide argmax (max value, ties -> lowest index).
#pragma unroll
  for (int off = 16; off >= 1; off >>= 1) {
    const float ov = __shfl_xor(bv, off, 32);
    const int   oi = __shfl_xor(bi, off, 32);
    if (ov > bv || (ov == bv && oi < bi)) { bv = ov; bi = oi; }
  }

  // Sum of exp(x - M); padded lanes contribute exp(-inf)=0.
  // __expf -> v_exp_f32 (TRANS pipe) keeps VALU off the critical path.
  float s = 0.0f;
#pragma unroll
  for (int it = 0; it < 8; ++it) {
#pragma unroll
    for (int j = 0; j < 4; ++j)
      s += __expf(buf[it][j] - M);
  }
#pragma unroll
  for (int off = 16; off >= 1; off >>= 1)
    s += __shfl_xor(s, off, 32);

  const int   t      = y_true[row];
  const float x_true = rowp[t];              // one extra 4B load per row
  const float ce     = logf(s) - (x_true - M);
  const float pt     = expf(x_true - M) / s;
  const int   pred   = bi;

  float w = 1.0f;
  if (pred != t && has_link[t] != 0 && link_matrix[t * C_DIM + pred] != 0)
    w = ALPHA_LINK;

  const float om   = 1.0f - pt;
  const float loss = w * om * om * ce;       // GAMMA == 2

  if (lane == 0) row_loss[row] = loss;
}

// Deterministic mean of N_ROWS floats via WMMA.
// D = A x B + C with B == ones(4x16)  =>  sum(D over all m,n) = 16 * sum(A).
// Identity holds for ANY bijective A lane layout, so each chunk element only
// needs to land exactly once in the 2 A VGPRs (v2f per lane). Four independent
// accumulator chains avoid the WMMA->WMMA RAW delay.
__global__ void __launch_bounds__(256)
lfl_reduce(const float* __restrict__ row_loss, float* __restrict__ out) {
  __shared__ float partial[8];
  const int lane = threadIdx.x & 31;
  const int wave = threadIdx.x >> 5;

  v8f acc0 = {0.f, 0.f, 0.f, 0.f, 0.f, 0.f, 0.f, 0.f};
  v8f acc1 = acc0, acc2 = acc0, acc3 = acc0;
  v2f ones = {1.f, 1.f};

  // 131072 / 64 = 2048 chunks of 64 floats; 8 waves x 256 chunks each.
  const int chunk0 = wave * 256;
  for (int c = 0; c < 256; c += 4) {
    const float* p0 = row_loss + (size_t)(chunk0 + c + 0) * 64 + lane * 2;
    const float* p1 = row_loss + (size_t)(chunk0 + c + 1) * 64 + lane * 2;
    const float* p2 = row_loss + (size_t)(chunk0 + c + 2) * 64 + lane * 2;
    const float* p3 = row_loss + (size_t)(chunk0 + c + 3) * 64 + lane * 2;
    v2f a0 = *(const v2f*)p0;                // 8B-aligned
    v2f a1 = *(const v2f*)p1;
    v2f a2 = *(const v2f*)p2;
    v2f a3 = *(const v2f*)p3;
#if __has_builtin(__builtin_amdgcn_wmma_f32_16x16x4_f32)
    acc0 = __builtin_amdgcn_wmma_f32_16x16x4_f32(false, a0, false, ones,
                                                 (short)0, acc0, false, false);
    acc1 = __builtin_amdgcn_wmma_f32_16x16x4_f32(false, a1, false, ones,
                                                 (short)0, acc1, false, false);
    acc2 = __builtin_amdgcn_wmma_f32_16x16x4_f32(false, a2, false, ones,
                                                 (short)0, acc2, false, false);
    acc3 = __builtin_amdgcn_wmma_f32_16x16x4_f32(false, a3, false, ones,
                                                 (short)0, acc3, false, false);
#else
    // VALU fallback preserving the x16 replication semantics.
    acc0[0] += 16.0f * (a0[0] + a0[1]);
    acc1[0] += 16.0f * (a1[0] + a1[1]);
    acc2[0] += 16.0f * (a2[0] + a2[1]);
    acc3[0] += 16.0f * (a3[0] + a3[1]);
#endif
  }

  // Sum all 4 accumulators' 8 VGPRs, then across 32 lanes.
  v8f accA = acc0 + acc1;
  v8f accB = acc2 + acc3;
  v8f acc  = accA + accB;
  float s = acc[0] + acc[1] + acc[2] + acc[3] + acc[4] + acc[5] + acc[6] + acc[7];
#pragma unroll
  for (int off = 16; off >= 1; off >>= 1)
    s += __shfl_xor(s, off, 32);

  if (lane == 0) partial[wave] = s;
  __syncthreads();

  if (threadIdx.x == 0) {
    float tot = 0.0f;
    for (int i = 0; i < 8; ++i) tot += partial[i];   // fixed order: deterministic
    out[0] = tot * (1.0f / 16.0f) * (1.0f / (float)N_ROWS);
  }
}

extern "C" void kernel_launch(void* const* d_in, const int* in_sizes, int n_in,
                              void* d_out, int out_size, void* d_ws, size_t ws_size,
                              hipStream_t stream) {
  (void)in_sizes; (void)n_in; (void)out_size; (void)ws_size;
  const float* y_pred      = (const float*)d_in[0];
  const int*   y_true      = (const int*)d_in[1];
  const int*   link_matrix = (const int*)d_in[2];
  const int*   has_link    = (const int*)d_in[3];
  float*       row_loss    = (float*)d_ws;           // N_ROWS * 4 bytes
  float*       out         = (float*)d_out;

  lfl_rowloss<<<N_ROWS / 8, 256, 0, stream>>>(y_pred, y_true, link_matrix,
                                              has_link, row_loss);
  lfl_reduce<<<1, 256, 0, stream>>>(row_loss, out);
}